// GIN_37340445671820
// MI455X (gfx1250) — compile-verified
//
#include <hip/hip_runtime.h>

#define N_NODES 40000
#define N_EDGES 640000
#define D       128

typedef float v2f __attribute__((ext_vector_type(2)));
typedef float v8f __attribute__((ext_vector_type(8)));
typedef unsigned int v4u __attribute__((ext_vector_type(4)));
typedef int v4i __attribute__((ext_vector_type(4)));
typedef int v8i __attribute__((ext_vector_type(8)));

#if defined(__AMDGCN__) && __has_builtin(__builtin_amdgcn_tensor_load_to_lds) && \
    __has_builtin(__builtin_amdgcn_s_wait_tensorcnt)
#define USE_TDM 1
#else
#define USE_TDM 0
#endif

// ---------------------------------------------------------------------------
// CSR build: zero counters -> histogram(dst) -> exclusive scan -> placement.
// Integer atomics only, once per call; reused by both layers.
// ---------------------------------------------------------------------------

__global__ void zero_kernel(int* __restrict__ p, int n) {
    int i = blockIdx.x * blockDim.x + threadIdx.x;
    int stride = gridDim.x * blockDim.x;
    for (; i < n; i += stride) p[i] = 0;
}

__global__ void hist_kernel(const int* __restrict__ dstIdx, int* __restrict__ cnt) {
    int i = blockIdx.x * blockDim.x + threadIdx.x;
    int stride = gridDim.x * blockDim.x;
    for (int e = i; e < N_EDGES; e += stride)
        atomicAdd(&cnt[dstIdx[e]], 1);          // global_atomic_add_u32
}

__global__ __launch_bounds__(256)
void scan_kernel(const int* __restrict__ cnt, int* __restrict__ off) {
    __shared__ int part[256];
    __shared__ int base[256];
    const int CH = (N_NODES + 255) / 256;       // 157 per thread
    const int t  = threadIdx.x;
    const int lo = t * CH;
    const int hi = min(lo + CH, N_NODES);

    int s = 0;
    for (int i = lo; i < hi; ++i) s += cnt[i];
    part[t] = s;
    __syncthreads();

    if (t == 0) {
        int run = 0;
        for (int i = 0; i < 256; ++i) { base[i] = run; run += part[i]; }
        off[N_NODES] = run;                     // == N_EDGES
    }
    __syncthreads();

    int run = base[t];
    for (int i = lo; i < hi; ++i) { off[i] = run; run += cnt[i]; }
}

__global__ void build_perm_kernel(const int* __restrict__ dstIdx,
                                  const int* __restrict__ off,
                                  int* __restrict__ cur,
                                  int* __restrict__ perm) {
    int i = blockIdx.x * blockDim.x + threadIdx.x;
    int stride = gridDim.x * blockDim.x;
    for (int e = i; e < N_EDGES; e += stride) {
        const int d   = dstIdx[e];
        const int pos = off[d] + atomicAdd(&cur[d], 1);
        perm[pos] = e;
    }
}

// ---------------------------------------------------------------------------
// Per-layer aggregation as a pure GATHER (no fp atomics): one wave per node,
// 32 lanes x float4 = node's 128-float accumulator, initialized from x.
// Unrolled x2 for two independent perm->src->x load chains per iteration.
// ---------------------------------------------------------------------------
__global__ __launch_bounds__(256)
void gather_kernel(const float* __restrict__ x,
                   const int* __restrict__ srcIdx,
                   const float* __restrict__ ew,
                   const float* __restrict__ ewW,
                   const float* __restrict__ ewb,
                   const int* __restrict__ off,
                   const int* __restrict__ perm,
                   float* __restrict__ buf) {
    const int lane = threadIdx.x & 31;
    const int node = (blockIdx.x * blockDim.x + threadIdx.x) >> 5;  // 1 wave/node
    if (node >= N_NODES) return;

    const float4 w4 = reinterpret_cast<const float4*>(ewW)[lane];
    const float4 b4 = reinterpret_cast<const float4*>(ewb)[lane];
    const float4* __restrict__ X4 = reinterpret_cast<const float4*>(x);

    float4 acc = X4[node * (D / 4) + lane];

    const int beg = off[node];
    const int end = off[node + 1];
    int j = beg;
    for (; j + 2 <= end; j += 2) {
        const int   e0 = perm[j],     e1 = perm[j + 1];
        const int   s0 = srcIdx[e0],  s1 = srcIdx[e1];
        const float t0 = ew[e0],      t1 = ew[e1];
        const float4 xv0 = X4[s0 * (D / 4) + lane];
        const float4 xv1 = X4[s1 * (D / 4) + lane];
        acc.x += fmaxf(xv0.x + fmaf(t0, w4.x, b4.x), 0.0f)
               + fmaxf(xv1.x + fmaf(t1, w4.x, b4.x), 0.0f);
        acc.y += fmaxf(xv0.y + fmaf(t0, w4.y, b4.y), 0.0f)
               + fmaxf(xv1.y + fmaf(t1, w4.y, b4.y), 0.0f);
        acc.z += fmaxf(xv0.z + fmaf(t0, w4.z, b4.z), 0.0f)
               + fmaxf(xv1.z + fmaf(t1, w4.z, b4.z), 0.0f);
        acc.w += fmaxf(xv0.w + fmaf(t0, w4.w, b4.w), 0.0f)
               + fmaxf(xv1.w + fmaf(t1, w4.w, b4.w), 0.0f);
    }
    if (j < end) {
        const int   e = perm[j];
        const int   s = srcIdx[e];
        const float t = ew[e];
        const float4 xv = X4[s * (D / 4) + lane];
        acc.x += fmaxf(xv.x + fmaf(t, w4.x, b4.x), 0.0f);
        acc.y += fmaxf(xv.y + fmaf(t, w4.y, b4.y), 0.0f);
        acc.z += fmaxf(xv.z + fmaf(t, w4.z, b4.z), 0.0f);
        acc.w += fmaxf(xv.w + fmaf(t, w4.w, b4.w), 0.0f);
    }
    reinterpret_cast<float4*>(buf)[node * (D / 4) + lane] = acc;
}

// ---------------------------------------------------------------------------
// out[m,n] = relu( X[m,:] @ W[:,n] + bias[n] ) with V_WMMA_F32_16X16X4_F32.
// Block = 8 waves; block owns 16 rows, wave w owns cols [16w,16w+16).
// W (64KB fp32) staged into LDS via TENSOR_LOAD_TO_LDS (TDM DMA) when the
// builtin is available; wave 0 issues the DMA, waits on TENSORcnt, then the
// workgroup barrier publishes the tile.
// ---------------------------------------------------------------------------
__global__ __launch_bounds__(256)
void gemm_bias_relu_kernel(const float* __restrict__ X,
                           const float* __restrict__ W,
                           const float* __restrict__ bias,
                           float* __restrict__ out) {
    __shared__ float lw[D * D];  // 64 KB

#if USE_TDM
    if (threadIdx.x < 32) {      // wave 0 only (EXEC all-ones for the wave)
        const unsigned long long ga = (unsigned long long)(uintptr_t)W;
        const unsigned int ldsOff   = (unsigned int)(uintptr_t)(&lw[0]);
        // D# group 0: count=1 | lds_addr | global_addr[56:0] | type=2
        v4u g0 = { 1u,
                   ldsOff,
                   (unsigned int)(ga & 0xFFFFFFFFu),
                   (unsigned int)((ga >> 32) & 0x01FFFFFFu) | 0x80000000u };
        // D# group 1: data_size=4B; tensor 128x128, tile 128x128, stride0=128
        v8i g1 = { (int)(2u << 16),          // [17:16] data_size=2 (4B)
                   (int)(128u << 16),        // [79:48] tensor_dim0=128 (low 16)
                   (int)(128u << 16),        // dim0 hi=0 | [111:80] tensor_dim1=128
                   (int)(128u << 16),        // dim1 hi=0 | [127:112] tile_dim0=128
                   (int)128,                 // [143:128] tile_dim1=128, tile_dim2=0
                   (int)128,                 // [207:160] tensor_dim0_stride=128
                   (int)(128u << 16),        // stride0 hi=0 | tensor_dim1_stride lo
                   0 };
        v4i gz = { 0, 0, 0, 0 };             // groups 2/3: dims unused (2D tile)
#if __clang_major__ >= 23
        v8i gz8 = { 0, 0, 0, 0, 0, 0, 0, 0 };
        __builtin_amdgcn_tensor_load_to_lds(g0, g1, gz, gz, gz8, 0);
#else
        __builtin_amdgcn_tensor_load_to_lds(g0, g1, gz, gz, 0);
#endif
        __builtin_amdgcn_s_wait_tensorcnt(0);
    }
#else
    for (int i = threadIdx.x; i < (D * D) / 4; i += blockDim.x) {
        reinterpret_cast<float4*>(lw)[i] = reinterpret_cast<const float4*>(W)[i];
    }
#endif
    __syncthreads();

    const int m0     = blockIdx.x * 16;
    const int waveid = threadIdx.x >> 5;
    const int lane   = threadIdx.x & 31;
    const int n0     = waveid * 16;
    const int hi     = lane >> 4;
    const int l15    = lane & 15;

    const float* arow = X + (size_t)(m0 + l15) * D;

    v8f acc = {};
    #pragma unroll 4
    for (int k = 0; k < D; k += 4) {
        // A 16x4 fp32: lane holds A[m][k+2*hi], A[m][k+2*hi+1] (contiguous)
        v2f a = *reinterpret_cast<const v2f*>(arow + k + 2 * hi);
        // B 4x16 fp32: VGPR g holds B[k + g + 2*hi][n0 + l15]
        v2f b;
        b.x = lw[(k + 2 * hi + 0) * D + n0 + l15];
        b.y = lw[(k + 2 * hi + 1) * D + n0 + l15];
        acc = __builtin_amdgcn_wmma_f32_16x16x4_f32(
                  false, a, false, b, (short)0, acc, false, false);
    }

    const float bv = bias[n0 + l15];
    #pragma unroll
    for (int r = 0; r < 8; ++r) {
        const int row = m0 + r + 8 * hi;   // C/D: VGPR r -> row r + 8*hi
        out[(size_t)row * D + n0 + l15] = fmaxf(acc[r] + bv, 0.0f);
    }
}

extern "C" void kernel_launch(void* const* d_in, const int* in_sizes, int n_in,
                              void* d_out, int out_size, void* d_ws, size_t ws_size,
                              hipStream_t stream) {
    const float* x     = (const float*)d_in[0];
    const int*   ei    = (const int*)d_in[1];   // (2, N_EDGES) row-major
    const float* ew    = (const float*)d_in[2];
    const float* ew1W  = (const float*)d_in[3];
    const float* ew1b  = (const float*)d_in[4];
    const float* nn1W  = (const float*)d_in[5];
    const float* nn1b  = (const float*)d_in[6];
    const float* ew2W  = (const float*)d_in[7];
    const float* ew2b  = (const float*)d_in[8];
    const float* nn2W  = (const float*)d_in[9];
    const float* nn2b  = (const float*)d_in[10];

    float* out = (float*)d_out;

    // Workspace layout (23.5 MB total):
    float* buf  = (float*)d_ws;                                   // 20,480,000 B
    int*   cnt  = (int*)((char*)d_ws + (size_t)N_NODES * D * 4);  // 40000
    int*   cur  = cnt + N_NODES;                                  // 40000
    int*   off  = cur + N_NODES;                                  // 40001
    int*   perm = off + (N_NODES + 1);                            // 640000

    const int* srcIdx = ei;
    const int* dstIdx = ei + N_EDGES;

    // ---- CSR build (once; shared by both layers) ----
    zero_kernel<<<313, 256, 0, stream>>>(cnt, 2 * N_NODES);       // cnt + cur
    hist_kernel<<<1024, 256, 0, stream>>>(dstIdx, cnt);
    scan_kernel<<<1, 256, 0, stream>>>(cnt, off);
    build_perm_kernel<<<1024, 256, 0, stream>>>(dstIdx, off, cur, perm);

    // ---- Layer 1 ----
    gather_kernel<<<(N_NODES * 32 + 255) / 256, 256, 0, stream>>>(
        x, srcIdx, ew, ew1W, ew1b, off, perm, buf);
    gemm_bias_relu_kernel<<<N_NODES / 16, 256, 0, stream>>>(buf, nn1W, nn1b, out);

    // ---- Layer 2 (h lives in d_out; stream order makes reuse safe) ----
    gather_kernel<<<(N_NODES * 32 + 255) / 256, 256, 0, stream>>>(
        out, srcIdx, ew, ew2W, ew2b, off, perm, buf);
    gemm_bias_relu_kernel<<<N_NODES / 16, 256, 0, stream>>>(buf, nn2W, nn2b, out);
}